// RPNClassificationLoss_8899172238095
// MI455X (gfx1250) — compile-verified
//
#include <hip/hip_runtime.h>
#include <math.h>
#include <stdint.h>

// ---------------- problem constants (match reference) ----------------
#define P_N   100000   // NUM_PROPOSALS
#define G_N   512      // NUM_GT
#define HALF  128      // N_CLS / 2
#define EPS_F 1e-7f

#define TPB1  256      // kernel 1 block size (8 wave32)
#define PPT   2        // proposals per thread (register blocking)
#define TPB2  1024     // kernel 2 single-block size

// GCC-style int4 vector to exactly match the async-LDS builtin's pointee type
typedef int v4i_t __attribute__((vector_size(4 * sizeof(int))));
#define AS1 __attribute__((address_space(1)))
#define AS3 __attribute__((address_space(3)))

// =====================================================================
// Kernel 1: max IoU per proposal.
// GT boxes (8KB) staged into LDS with gfx1250 async global->LDS copy
// (ASYNCcnt-tracked), then every lane walks the 512 GT boxes via LDS
// broadcast reads, updating 2 proposals' running max IoU in registers.
// =====================================================================
__global__ __launch_bounds__(TPB1) void rpn_maxiou_kernel(
    const float4* __restrict__ proposals4,   // (P,4) xyxy
    const float4* __restrict__ gt4,          // (G,4) xyxy
    float* __restrict__ out_maxiou)          // (P,)
{
    __shared__ float4 sGT[G_N];    // 8 KB
    __shared__ float  sArea[G_N];  // 2 KB

    const int t = threadIdx.x;

    // ---- stage GT boxes into LDS (async on gfx1250) ----
#if __has_builtin(__builtin_amdgcn_global_load_async_to_lds_b128)
    for (int i = t; i < G_N; i += TPB1) {
        AS1 v4i_t* gsrc = (AS1 v4i_t*)(uintptr_t)(const void*)(gt4 + i);
        AS3 v4i_t* ldst = (AS3 v4i_t*)(uint32_t)(uintptr_t)(void*)(&sGT[i]);
        __builtin_amdgcn_global_load_async_to_lds_b128(gsrc, ldst,
                                                       /*imm offset*/ 0,
                                                       /*cpol*/ 0);
    }
#if __has_builtin(__builtin_amdgcn_s_wait_asynccnt)
    __builtin_amdgcn_s_wait_asynccnt(0);
#else
    asm volatile("s_wait_asynccnt 0" ::: "memory");
#endif
#else
    for (int i = t; i < G_N; i += TPB1) sGT[i] = gt4[i];
#endif
    __syncthreads();

    // ---- precompute GT areas in LDS ----
    for (int i = t; i < G_N; i += TPB1) {
        float4 b = sGT[i];
        sArea[i] = (b.z - b.x) * (b.w - b.y);
    }
    __syncthreads();

    // ---- register-load PPT proposals per lane ----
    const int idx0 = (blockIdx.x * TPB1 + t) * PPT;
    float4 pb[PPT];
    float  pa[PPT];
    float  m[PPT];
#pragma unroll
    for (int j = 0; j < PPT; ++j) {
        int idx = idx0 + j;
        float4 b = proposals4[idx < P_N ? idx : 0];
        pb[j] = b;
        pa[j] = (b.z - b.x) * (b.w - b.y);
        m[j]  = 0.0f;   // IoU >= 0 always (intersection clamped at 0)
    }

    // ---- sweep GT boxes: LDS broadcast read, PPT IoUs per box ----
    for (int g = 0; g < G_N; ++g) {
        float4 gb = sGT[g];
        float  ga = sArea[g];
#pragma unroll
        for (int j = 0; j < PPT; ++j) {
            float lx = fmaxf(gb.x, pb[j].x);
            float ly = fmaxf(gb.y, pb[j].y);
            float rx = fminf(gb.z, pb[j].z);
            float ry = fminf(gb.w, pb[j].w);
            float w  = fmaxf(rx - lx, 0.0f);
            float h  = fmaxf(ry - ly, 0.0f);
            float inter = w * h;
            float denom = ga + pa[j] - inter;
            float iou   = inter * __builtin_amdgcn_rcpf(denom);
            m[j] = fmaxf(m[j], iou);
        }
    }

#pragma unroll
    for (int j = 0; j < PPT; ++j) {
        int idx = idx0 + j;
        if (idx < P_N) out_maxiou[idx] = m[j];
    }
}

// =====================================================================
// Kernel 2: deterministic top-128 / bottom-128 selection + BCE mean.
// Keys are unique 48-bit values: (float_bits << 17) | tie_break(index)
// (max_iou >= 0 so float bits are order-preserving; P < 2^17).
// 8-bit MSD radix-select: 6 histogram passes per search (vs 47 for a
// bitwise binary search) drills down to the exact key of the 128th
// largest / smallest element. All counting uses integer LDS atomics
// (order-independent) and the final float sum uses a tree reduction,
// so the result is bit-deterministic across replays.
// =====================================================================
__global__ __launch_bounds__(TPB2) void rpn_topk_bce_kernel(
    const float* __restrict__ maxiou,   // (P,)
    const float* __restrict__ obj,      // (P,) probabilities
    float* __restrict__ out)            // (1,)
{
    __shared__ int   s_bins[256];
    __shared__ float s_red[TPB2];

    const int t = threadIdx.x;

    // key helpers
    auto keyTop = [&](int i) -> unsigned long long {
        unsigned int b = __float_as_uint(maxiou[i]);
        return (((unsigned long long)b) << 17) |
               (unsigned long long)(0x1FFFFu - (unsigned)i);  // ties: low index wins
    };
    auto keyBot = [&](int i) -> unsigned long long {
        unsigned int b = __float_as_uint(maxiou[i]);
        return (((unsigned long long)b) << 17) | (unsigned long long)(unsigned)i;
    };

    // ---- radix-select the exact key of the 128th-LARGEST keyTop ----
    unsigned long long prefixTop = 0ull;
    {
        int kk = HALF;  // rank of target within keys matching prefix
        for (int shift = 40; shift >= 0; shift -= 8) {
            if (t < 256) s_bins[t] = 0;
            __syncthreads();
            for (int i = t; i < P_N; i += TPB2) {
                unsigned long long key = keyTop(i);
                if ((key >> (shift + 8)) == prefixTop)
                    atomicAdd(&s_bins[(int)((key >> shift) & 0xFFull)], 1);
            }
            __syncthreads();
            // uniform redundant scan from the top bin down
            int cum = 0, b = 0;
            for (int j = 255; j >= 0; --j) {
                int c = s_bins[j];
                if (cum + c >= kk) { b = j; break; }
                cum += c;
            }
            kk -= cum;
            prefixTop = (prefixTop << 8) | (unsigned long long)(unsigned)b;
            __syncthreads();
        }
    }
    const unsigned long long Tpos = prefixTop;  // select: keyTop >= Tpos (exactly 128)

    // ---- radix-select the exact key of the 128th-SMALLEST keyBot ----
    unsigned long long prefixBot = 0ull;
    {
        int kk = HALF;
        for (int shift = 40; shift >= 0; shift -= 8) {
            if (t < 256) s_bins[t] = 0;
            __syncthreads();
            for (int i = t; i < P_N; i += TPB2) {
                unsigned long long key = keyBot(i);
                if ((key >> (shift + 8)) == prefixBot)
                    atomicAdd(&s_bins[(int)((key >> shift) & 0xFFull)], 1);
            }
            __syncthreads();
            // uniform redundant scan from the bottom bin up
            int cum = 0, b = 0;
            for (int j = 0; j < 256; ++j) {
                int c = s_bins[j];
                if (cum + c >= kk) { b = j; break; }
                cum += c;
            }
            kk -= cum;
            prefixBot = (prefixBot << 8) | (unsigned long long)(unsigned)b;
            __syncthreads();
        }
    }
    const unsigned long long Tneg = prefixBot;  // select: keyBot <= Tneg (exactly 128)

    // ---- accumulate BCE terms over the exactly-128 + exactly-128 selected ----
    float acc = 0.0f;
    for (int i = t; i < P_N; i += TPB2) {
        unsigned long long kt = keyTop(i);
        unsigned long long kb = keyBot(i);
        bool selPos = (kt >= Tpos);
        bool selNeg = (kb <= Tneg);
        if (selPos || selNeg) {
            float v = maxiou[i];
            float p = fminf(fmaxf(obj[i], EPS_F), 1.0f - EPS_F);
            if (selPos) acc += (v > 0.5f) ? logf(p) : log1pf(-p); // label = iou > 0.5
            if (selNeg) acc += log1pf(-p);                        // label = 0
        }
    }

    // deterministic tree reduction
    s_red[t] = acc;
    __syncthreads();
    for (int s = TPB2 / 2; s > 0; s >>= 1) {
        if (t < s) s_red[t] += s_red[t + s];
        __syncthreads();
    }
    if (t == 0) out[0] = -s_red[0] * (1.0f / (2.0f * HALF));
}

// =====================================================================
// Launch wrapper
// =====================================================================
extern "C" void kernel_launch(void* const* d_in, const int* in_sizes, int n_in,
                              void* d_out, int out_size, void* d_ws, size_t ws_size,
                              hipStream_t stream) {
    (void)in_sizes; (void)n_in; (void)out_size; (void)ws_size;

    const float* obj       = (const float*)d_in[0];   // (P,)
    const float* proposals = (const float*)d_in[1];   // (P,4)
    const float* gt        = (const float*)d_in[2];   // (G,4)
    float*       out       = (float*)d_out;           // (1,)
    float*       maxiou    = (float*)d_ws;            // P floats = 400 KB scratch

    const int threads_needed = (P_N + PPT - 1) / PPT;              // 50000
    const int blocks1        = (threads_needed + TPB1 - 1) / TPB1; // 196

    rpn_maxiou_kernel<<<dim3(blocks1), dim3(TPB1), 0, stream>>>(
        (const float4*)proposals, (const float4*)gt, maxiou);

    rpn_topk_bce_kernel<<<dim3(1), dim3(TPB2), 0, stream>>>(maxiou, obj, out);
}